// LightGCN_86887188398691
// MI455X (gfx1250) — compile-verified
//
#include <hip/hip_runtime.h>

#define USER_NUM 100000
#define ITEM_NUM 50000
#define N_TOTAL  (USER_NUM + ITEM_NUM)
#define REC_DIM  256
#define CHUNKS   (REC_DIM / 4)   // float4 chunks per row

typedef __attribute__((ext_vector_type(2))) float v2f;
typedef __attribute__((ext_vector_type(8))) float v8f;

// cur[N][D] = concat(user_emb, item_emb); exact grid, no guard needed.
__global__ void lgcn_concat(const float4* __restrict__ ue,
                            const float4* __restrict__ ie,
                            float4* __restrict__ cur) {
  size_t t = (size_t)blockIdx.x * blockDim.x + threadIdx.x;
  const size_t usz = (size_t)USER_NUM * CHUNKS;
  cur[t] = (t < usz) ? ue[t] : ie[t - usz];
}

// uacc[b] = user_emb[users[b]] ; iacc[b] = item_emb[items[b]]  (layer-0 term)
__global__ void lgcn_init_batch(const float4* __restrict__ ue,
                                const float4* __restrict__ ie,
                                const int* __restrict__ users,
                                const int* __restrict__ items,
                                float4* __restrict__ uacc,
                                float4* __restrict__ iacc) {
  size_t t = (size_t)blockIdx.x * blockDim.x + threadIdx.x;
  int b = (int)(t >> 6);
  int c = (int)(t & 63);
  uacc[t] = ue[(size_t)users[b] * CHUNKS + c];
  iacc[t] = ie[(size_t)items[b] * CHUNKS + c];
}

// One wave per edge: lane handles dims [4*lane,4*lane+4) and [128+4*lane, ...).
// Gather b128 loads hit the 192MB L2 (cur fits entirely); scatter via
// no-return agent-scope fp32 atomics. Edge arrays streamed non-temporally.
__global__ void __launch_bounds__(256) lgcn_spmm(
    const float* __restrict__ cur, float* __restrict__ nxt,
    const int* __restrict__ rows, const int* __restrict__ cols,
    const float* __restrict__ vals, int nnz) {
  int wave = threadIdx.x >> 5;
  int lane = threadIdx.x & 31;
  int e = blockIdx.x * 8 + wave;
  if (e >= nnz) return;

  int r  = __builtin_nontemporal_load(rows + e);
  int cl = __builtin_nontemporal_load(cols + e);
  float v = __builtin_nontemporal_load(vals + e);

  const float4* src = (const float4*)(cur + (size_t)cl * REC_DIM);
  float4 x0 = src[lane];
  float4 x1 = src[lane + 32];

  float* dst = nxt + (size_t)r * REC_DIM + 4 * lane;
  __hip_atomic_fetch_add(dst + 0,   v * x0.x, __ATOMIC_RELAXED, __HIP_MEMORY_SCOPE_AGENT);
  __hip_atomic_fetch_add(dst + 1,   v * x0.y, __ATOMIC_RELAXED, __HIP_MEMORY_SCOPE_AGENT);
  __hip_atomic_fetch_add(dst + 2,   v * x0.z, __ATOMIC_RELAXED, __HIP_MEMORY_SCOPE_AGENT);
  __hip_atomic_fetch_add(dst + 3,   v * x0.w, __ATOMIC_RELAXED, __HIP_MEMORY_SCOPE_AGENT);
  __hip_atomic_fetch_add(dst + 128, v * x1.x, __ATOMIC_RELAXED, __HIP_MEMORY_SCOPE_AGENT);
  __hip_atomic_fetch_add(dst + 129, v * x1.y, __ATOMIC_RELAXED, __HIP_MEMORY_SCOPE_AGENT);
  __hip_atomic_fetch_add(dst + 130, v * x1.z, __ATOMIC_RELAXED, __HIP_MEMORY_SCOPE_AGENT);
  __hip_atomic_fetch_add(dst + 131, v * x1.w, __ATOMIC_RELAXED, __HIP_MEMORY_SCOPE_AGENT);
}

// After each layer: uacc[b] += layer[users[b]] ; iacc[b] += layer[USER_NUM+items[b]]
__global__ void lgcn_accum_batch(const float4* __restrict__ layer,
                                 const int* __restrict__ users,
                                 const int* __restrict__ items,
                                 float4* __restrict__ uacc,
                                 float4* __restrict__ iacc) {
  size_t t = (size_t)blockIdx.x * blockDim.x + threadIdx.x;
  int b = (int)(t >> 6);
  int c = (int)(t & 63);
  float4 xu = layer[(size_t)users[b] * CHUNKS + c];
  float4 xi = layer[(size_t)(USER_NUM + items[b]) * CHUNKS + c];
  float4 au = uacc[t];
  float4 ai = iacc[t];
  au.x += xu.x; au.y += xu.y; au.z += xu.z; au.w += xu.w;
  ai.x += xi.x; ai.y += xi.y; ai.z += xi.z; ai.w += xi.w;
  uacc[t] = au;
  iacc[t] = ai;
}

// Batched dot via V_WMMA_F32_16X16X4_F32: wave handles 16 pairs, K=256 in 64
// steps; gamma = diag(U * It^T) / 16 (mean scaling 1/4 per factor).
// A layout (16x4 f32): lanes 0-15 hold {k,k+1}, lanes 16-31 hold {k+2,k+3} of
// row M = lane&15. B = It^T (4x16) has the symmetric per-lane layout.
// Diagonal of C/D (16x16 f32, 8 VGPRs): m<8 -> VGPR m @ lane m; m>=8 ->
// VGPR m-8 @ lane m+16.
__global__ void __launch_bounds__(256) lgcn_wmma_dot(
    const float* __restrict__ uacc, const float* __restrict__ iacc,
    float* __restrict__ gamma) {
  int lane = threadIdx.x & 31;
  int wave = threadIdx.x >> 5;
  int base = (blockIdx.x * 8 + wave) * 16;
  int m    = lane & 15;
  int half = lane >> 4;

  const float* up = uacc + (size_t)(base + m) * REC_DIM + 2 * half;
  const float* ip = iacc + (size_t)(base + m) * REC_DIM + 2 * half;

  v8f c = {0.f, 0.f, 0.f, 0.f, 0.f, 0.f, 0.f, 0.f};
#pragma unroll 8
  for (int k = 0; k < REC_DIM; k += 4) {
    v2f a = *(const v2f*)(up + k);
    v2f b = *(const v2f*)(ip + k);
    c = __builtin_amdgcn_wmma_f32_16x16x4_f32(false, a, false, b,
                                              (short)0, c, false, false);
  }

  float dv[8];
#pragma unroll
  for (int i = 0; i < 8; ++i) dv[i] = c[i];

  if (lane < 8) {
    gamma[base + lane] = dv[lane] * 0.0625f;
  } else if (lane >= 24) {
    gamma[base + lane - 16] = dv[lane - 24] * 0.0625f;
  }
}

extern "C" void kernel_launch(void* const* d_in, const int* in_sizes, int n_in,
                              void* d_out, int out_size, void* d_ws, size_t ws_size,
                              hipStream_t stream) {
  const float* user_emb = (const float*)d_in[0];
  const float* item_emb = (const float*)d_in[1];
  const float* vals     = (const float*)d_in[2];
  const int*   rows     = (const int*)d_in[3];
  const int*   cols     = (const int*)d_in[4];
  const int*   users    = (const int*)d_in[5];
  const int*   items    = (const int*)d_in[6];
  float* gamma = (float*)d_out;

  const int nnz   = in_sizes[2];
  const int batch = in_sizes[5];

  float* cur  = (float*)d_ws;                       // N*256 f32
  float* nxt  = cur  + (size_t)N_TOTAL * REC_DIM;   // N*256 f32
  float* uacc = nxt  + (size_t)N_TOTAL * REC_DIM;   // batch*256 f32
  float* iacc = uacc + (size_t)batch   * REC_DIM;   // batch*256 f32

  // Layer 0: cur = concat(emb), accumulators = gathered layer-0 rows.
  lgcn_concat<<<(N_TOTAL * CHUNKS) / 256, 256, 0, stream>>>(
      (const float4*)user_emb, (const float4*)item_emb, (float4*)cur);
  lgcn_init_batch<<<(batch * CHUNKS) / 256, 256, 0, stream>>>(
      (const float4*)user_emb, (const float4*)item_emb, users, items,
      (float4*)uacc, (float4*)iacc);

  for (int layer = 0; layer < 3; ++layer) {
    hipMemsetAsync(nxt, 0, (size_t)N_TOTAL * REC_DIM * sizeof(float), stream);
    lgcn_spmm<<<(nnz + 7) / 8, 256, 0, stream>>>(cur, nxt, rows, cols, vals, nnz);
    lgcn_accum_batch<<<(batch * CHUNKS) / 256, 256, 0, stream>>>(
        (const float4*)nxt, users, items, (float4*)uacc, (float4*)iacc);
    float* tmp = cur; cur = nxt; nxt = tmp;
  }

  // gamma[b] = dot(uacc[b], iacc[b]) / 16 via f32 WMMA diagonal.
  lgcn_wmma_dot<<<batch / 128, 256, 0, stream>>>(uacc, iacc, gamma);
}